// LILayer_19816979103824
// MI455X (gfx1250) — compile-verified
//
#include <hip/hip_runtime.h>
#include <hip/hip_bf16.h>

// Problem constants (match reference)
constexpr int B_   = 8;
constexpr int T_   = 2048;
constexpr int IN_  = 1024;
constexpr int OUT_ = 512;
constexpr int M_   = B_ * T_;          // 16384 rows of the GEMM
constexpr int CH_  = B_ * OUT_;        // 4096 scan channels
constexpr int NC_  = 16;               // scan chunks
constexpr int CL_  = T_ / NC_;         // 128 timesteps per chunk

typedef __attribute__((ext_vector_type(2))) float v2f;
typedef __attribute__((ext_vector_type(8))) float v8f;

// Register tiling: each wave computes a 64x64 C macro-tile
constexpr int WM = 4;                  // 16-row subtiles per wave (M)
constexpr int WN = 4;                  // 16-col subtiles per wave (N)
// 8 waves / block arranged 2(M) x 4(N) -> block tile 128 x 256
constexpr int BLK_M = 2 * WM * 16;     // 128
constexpr int BLK_N = 4 * WN * 16;     // 256

// ---------------------------------------------------------------------------
// GEMM: C[m,n] = sum_k X[m,k] * W[n,k] + bias[n]
// V_WMMA_F32_16X16X4_F32, 16 accumulators per wave (4Mx4N subtiles).
// Per K-step of 4: 4 A-fragment + 4 B-fragment b64 loads feed 16 WMMAs
// (loads:wmma = 0.5). Fragment traffic ~128 B/wmma.
// A fragment (16x4 f32): lanes 0-15 -> M=lane, K={k,k+1}; lanes 16-31 ->
// M=lane-16, K={k+2,k+3}: one contiguous float2 per lane. B (4x16) mirrors
// with N across lanes; W is row-major in K, so also one float2 per lane.
// ---------------------------------------------------------------------------
__global__ __launch_bounds__(256) void li_gemm_wmma_f32(
    const float* __restrict__ X,     // [M_, IN_]
    const float* __restrict__ W,     // [OUT_, IN_]
    const float* __restrict__ bias,  // [OUT_]
    float* __restrict__ C)           // [M_, OUT_]
{
    const int lane  = threadIdx.x & 31;
    const int wave  = threadIdx.x >> 5;
    const int waveM = wave & 1;            // 2 waves along M
    const int waveN = wave >> 1;           // 4 waves along N
    const int mBase = blockIdx.x * BLK_M + waveM * (WM * 16);
    const int nBase = blockIdx.y * BLK_N + waveN * (WN * 16);

    const int l15   = lane & 15;
    const int khalf = (lane >> 4) << 1;    // 0 or 2

    const float* __restrict__ xp = X + (size_t)(mBase + l15) * IN_ + khalf;
    const float* __restrict__ wp = W + (size_t)(nBase + l15) * IN_ + khalf;

    v8f acc[WM][WN];
#pragma unroll
    for (int i = 0; i < WM; ++i)
#pragma unroll
        for (int j = 0; j < WN; ++j)
            acc[i][j] = (v8f){};

    for (int k = 0; k < IN_; k += 4) {
        v2f a[WM];
        v2f b[WN];
#pragma unroll
        for (int i = 0; i < WM; ++i)   // subtile offsets are 24-bit immediates
            a[i] = *(const v2f*)(xp + (size_t)i * 16 * IN_ + k);
#pragma unroll
        for (int j = 0; j < WN; ++j)
            b[j] = *(const v2f*)(wp + (size_t)j * 16 * IN_ + k);
#pragma unroll
        for (int i = 0; i < WM; ++i)
#pragma unroll
            for (int j = 0; j < WN; ++j)
                acc[i][j] = __builtin_amdgcn_wmma_f32_16x16x4_f32(
                    /*neg_a=*/false, a[i], /*neg_b=*/false, b[j],
                    /*c_mod=*/(short)0, acc[i][j],
                    /*reuse_a=*/false, /*reuse_b=*/false);
    }

    // D layout: VGPR r -> M = r (lanes 0-15) / M = 8+r (lanes 16-31), N = lane&15
    const int mHalf = (lane >> 4) << 3;    // 0 or 8
#pragma unroll
    for (int j = 0; j < WN; ++j) {
        const int n    = nBase + j * 16 + l15;
        const float bv = bias[n];
#pragma unroll
        for (int i = 0; i < WM; ++i) {
            const int mOff = mBase + i * 16 + mHalf;
#pragma unroll
            for (int r = 0; r < 8; ++r) {
                C[(size_t)(mOff + r) * OUT_ + n] = acc[i][j][r] + bv;
            }
        }
    }
}

// ---------------------------------------------------------------------------
// Blocked scan, pass 1: per (channel, chunk) local scan with v=0, store the
// chunk's final value (carry contribution).
// ---------------------------------------------------------------------------
__global__ __launch_bounds__(256) void li_scan_pass1(
    const float* __restrict__ C,       // [B_, T_, OUT_] (GEMM result)
    const float* __restrict__ decay,   // [OUT_]
    float* __restrict__ carry)         // [NC_, CH_]
{
    const int tid = blockIdx.x * blockDim.x + threadIdx.x;
    const int ch  = tid & (CH_ - 1);       // fast index: consecutive o -> coalesced
    const int j   = tid >> 12;             // chunk (CH_ = 2^12)
    const int b   = ch >> 9;               // / OUT_ (512 = 2^9)
    const int o   = ch & (OUT_ - 1);

    const float d  = decay[o];
    const float om = 1.0f - d;
    const float* __restrict__ p = C + ((size_t)b * T_ + (size_t)j * CL_) * OUT_ + o;

    float v = 0.0f;
    for (int t = 0; t < CL_; ++t) {
        v = d * v + om * p[(size_t)t * OUT_];
    }
    carry[(size_t)j * CH_ + ch] = v;
}

// ---------------------------------------------------------------------------
// Pass 2: per channel, sequentially combine the NC_ chunk carries to get the
// incoming state for each chunk:  vin[j] = v before chunk j.
// ---------------------------------------------------------------------------
__global__ __launch_bounds__(256) void li_scan_pass2(
    const float* __restrict__ decay,   // [OUT_]
    const float* __restrict__ carry,   // [NC_, CH_]
    float* __restrict__ vin)           // [NC_, CH_]
{
    const int ch = blockIdx.x * blockDim.x + threadIdx.x;
    const int o  = ch & (OUT_ - 1);
    const float d = decay[o];

    // dL = d^CL_ (CL_ = 128): 7 squarings
    float dL = d;
#pragma unroll
    for (int i = 0; i < 7; ++i) dL = dL * dL;

    float v = 0.0f;
    for (int j = 0; j < NC_; ++j) {
        vin[(size_t)j * CH_ + ch] = v;
        v = dL * v + carry[(size_t)j * CH_ + ch];
    }
}

// ---------------------------------------------------------------------------
// Pass 3: re-run the local scan seeded with the correct incoming state.
// Overwrites C in place (it becomes `outputs`) and mirrors into `states`
// (states[0,b,t+1,:] = outputs[b,t,:]; states[0,b,0,:] = 0).
// ---------------------------------------------------------------------------
__global__ __launch_bounds__(256) void li_scan_pass3(
    float* __restrict__ C,             // in: current, out: outputs  [B_,T_,OUT_]
    const float* __restrict__ decay,   // [OUT_]
    const float* __restrict__ vin,     // [NC_, CH_]
    float* __restrict__ states)        // [B_, T_+1, OUT_]
{
    const int tid = blockIdx.x * blockDim.x + threadIdx.x;
    const int ch  = tid & (CH_ - 1);
    const int j   = tid >> 12;
    const int b   = ch >> 9;
    const int o   = ch & (OUT_ - 1);

    const float d  = decay[o];
    const float om = 1.0f - d;

    float* __restrict__ p = C + ((size_t)b * T_ + (size_t)j * CL_) * OUT_ + o;
    float* __restrict__ s = states + ((size_t)b * (T_ + 1) + (size_t)j * CL_ + 1) * OUT_ + o;

    if (j == 0) {
        states[((size_t)b * (T_ + 1)) * OUT_ + o] = 0.0f;   // v0 row
    }

    float v = vin[(size_t)j * CH_ + ch];
    for (int t = 0; t < CL_; ++t) {
        v = d * v + om * p[(size_t)t * OUT_];   // read c_t
        p[(size_t)t * OUT_] = v;                // overwrite with v_t (outputs)
        s[(size_t)t * OUT_] = v;                // states[0,b,t+1,:]
    }
}

// ---------------------------------------------------------------------------
// kernel_launch
//   d_in[0] = input_tensor [B,T,IN]  f32
//   d_in[1] = weight       [OUT,IN]  f32
//   d_in[2] = bias         [OUT]     f32
//   d_in[3] = decay        [OUT]     f32
//   d_out   = outputs [B,T,OUT] ++ states [1,B,T+1,OUT]   (f32, flat)
// ---------------------------------------------------------------------------
extern "C" void kernel_launch(void* const* d_in, const int* in_sizes, int n_in,
                              void* d_out, int out_size, void* d_ws, size_t ws_size,
                              hipStream_t stream) {
    const float* X     = (const float*)d_in[0];
    const float* W     = (const float*)d_in[1];
    const float* bias  = (const float*)d_in[2];
    const float* decay = (const float*)d_in[3];

    float* outputs = (float*)d_out;                         // [B_,T_,OUT_]
    float* states  = outputs + (size_t)M_ * OUT_;           // [B_,T_+1,OUT_]

    float* carry = (float*)d_ws;                            // [NC_, CH_]
    float* vin   = carry + (size_t)NC_ * CH_;               // [NC_, CH_]

    // 1) GEMM (writes "current" into the outputs region)
    {
        dim3 grid(M_ / BLK_M, OUT_ / BLK_N);   // (128, 2)
        li_gemm_wmma_f32<<<grid, 256, 0, stream>>>(X, W, bias, outputs);
    }
    // 2) scan pass 1: local chunk scans -> carries
    {
        const int total = CH_ * NC_;     // 65536
        li_scan_pass1<<<total / 256, 256, 0, stream>>>(outputs, decay, carry);
    }
    // 3) scan pass 2: combine carries -> per-chunk incoming states
    {
        li_scan_pass2<<<CH_ / 256, 256, 0, stream>>>(decay, carry, vin);
    }
    // 4) scan pass 3: final scan, in-place outputs + states
    {
        const int total = CH_ * NC_;
        li_scan_pass3<<<total / 256, 256, 0, stream>>>(outputs, decay, vin, states);
    }
}